// GraphMETNetwork_simple_74826920231167
// MI455X (gfx1250) — compile-verified
//
#include <hip/hip_runtime.h>
#include <math.h>

typedef __attribute__((ext_vector_type(2))) float v2f;
typedef __attribute__((ext_vector_type(8))) float v8f;

#define IN_DIM 11
#define HID 32

// ---------------- degree / normalization ----------------

__global__ void deg_kernel(const int* __restrict__ dst, float* __restrict__ deg, int E) {
    int e = blockIdx.x * blockDim.x + threadIdx.x;
    if (e < E) atomicAdd(&deg[dst[e]], 1.0f);
}

__global__ void rsqrt_kernel(float* __restrict__ dinv, int n) {
    int i = blockIdx.x * blockDim.x + threadIdx.x;
    if (i < n) dinv[i] = rsqrtf(dinv[i] + 1.0f);   // +1 self-loop
}

// ---------------- WMMA GEMM: xw = x(n x 11) @ W1(11 x 32), K padded to 12 ----------------
// One wave32 per 16x16 output tile. V_WMMA_F32_16X16X4_F32, 3 K-steps.

__global__ void gemm_x_w1(const float* __restrict__ x, const float* __restrict__ W1,
                          float* __restrict__ out, int n) {
    int wid  = threadIdx.x >> 5;
    int tile = blockIdx.x * (blockDim.x >> 5) + wid;
    int mTiles = (n + 15) >> 4;
    if (tile >= mTiles * 2) return;          // wave-uniform branch (EXEC stays all-1 for WMMA)
    int tileN = tile & 1;
    int tileM = tile >> 1;
    int lane = threadIdx.x & 31;
    int half = lane >> 4;                    // lanes 16-31 hold K+2 (A) / upper K rows (B)
    int lm   = lane & 15;
    int row  = tileM * 16 + lm;
    int rowc = row < n ? row : n - 1;        // clamp pad rows (results masked at store)
    int col  = tileN * 16 + lm;

    v8f c = {0.f,0.f,0.f,0.f,0.f,0.f,0.f,0.f};
#pragma unroll
    for (int kk = 0; kk < 3; ++kk) {
        int k0 = kk * 4 + half * 2;
        v2f a, b;
        a.x = (k0     < IN_DIM) ? x[rowc * IN_DIM + k0    ] : 0.0f;
        a.y = (k0 + 1 < IN_DIM) ? x[rowc * IN_DIM + k0 + 1] : 0.0f;
        b.x = (k0     < IN_DIM) ? W1[(k0    ) * HID + col] : 0.0f;
        b.y = (k0 + 1 < IN_DIM) ? W1[(k0 + 1) * HID + col] : 0.0f;
        c = __builtin_amdgcn_wmma_f32_16x16x4_f32(false, a, false, b, (short)0, c, false, false);
    }
#pragma unroll
    for (int r = 0; r < 8; ++r) {
        int node = tileM * 16 + r + half * 8;   // D layout: VGPR r = rows r (lanes 0-15), r+8 (16-31)
        if (node < n) out[(size_t)node * HID + col] = c[r];
    }
}

// ---------------- WMMA GEMM: out = h(n x 32) @ W(32 x 32) ----------------

__global__ void gemm_h_w(const float* __restrict__ h, const float* __restrict__ W,
                         float* __restrict__ out, int n) {
    int wid  = threadIdx.x >> 5;
    int tile = blockIdx.x * (blockDim.x >> 5) + wid;
    int mTiles = (n + 15) >> 4;
    if (tile >= mTiles * 2) return;
    int tileN = tile & 1;
    int tileM = tile >> 1;
    int lane = threadIdx.x & 31;
    int half = lane >> 4;
    int lm   = lane & 15;
    int row  = tileM * 16 + lm;
    int rowc = row < n ? row : n - 1;
    int col  = tileN * 16 + lm;
    const float* arow = h + (size_t)rowc * HID;

    v8f c = {0.f,0.f,0.f,0.f,0.f,0.f,0.f,0.f};
#pragma unroll
    for (int kk = 0; kk < 8; ++kk) {
        int k0 = kk * 4 + half * 2;
        v2f a, b;
        a.x = arow[k0];
        a.y = arow[k0 + 1];
        b.x = W[(k0    ) * HID + col];
        b.y = W[(k0 + 1) * HID + col];
        c = __builtin_amdgcn_wmma_f32_16x16x4_f32(false, a, false, b, (short)0, c, false, false);
    }
#pragma unroll
    for (int r = 0; r < 8; ++r) {
        int node = tileM * 16 + r + half * 8;
        if (node < n) out[(size_t)node * HID + col] = c[r];
    }
}

// ---------------- edge scatter: agg[d] += dinv[s]*dinv[d]*feat[s]  (4 threads/edge) ----------------

__global__ void scatter_kernel(const int* __restrict__ src, const int* __restrict__ dst,
                               const float* __restrict__ dinv, const float* __restrict__ feat,
                               float* __restrict__ agg, int E) {
    int t = blockIdx.x * blockDim.x + threadIdx.x;
    int e = t >> 2;
    if (e >= E) return;
    int c0 = (t & 3) * 8;
    int s = src[e], d = dst[e];
    float w = dinv[s] * dinv[d];
    const float4* f = (const float4*)(feat + (size_t)s * HID + c0);
    float4 f0 = f[0];
    float4 f1 = f[1];
    float* a = agg + (size_t)d * HID + c0;
    atomicAdd(a + 0, w * f0.x);
    atomicAdd(a + 1, w * f0.y);
    atomicAdd(a + 2, w * f0.z);
    atomicAdd(a + 3, w * f0.w);
    atomicAdd(a + 4, w * f1.x);
    atomicAdd(a + 5, w * f1.y);
    atomicAdd(a + 6, w * f1.z);
    atomicAdd(a + 7, w * f1.w);
}

// ---------------- finalize: out = relu(agg + dinv^2 * feat + bias)  (self-loop fused) ----------------

__global__ void finalize_kernel(const float* __restrict__ agg, const float* feat,
                                const float* __restrict__ dinv, const float* __restrict__ bias,
                                float* out, int n) {
    int t = blockIdx.x * blockDim.x + threadIdx.x;
    if (t >= n * HID) return;
    int i = t >> 5;
    int c = t & 31;
    float di = dinv[i];
    float v = agg[t] + di * di * feat[t] + bias[c];
    out[t] = fmaxf(v, 0.0f);
}

// ---------------- output MLP: 32 -> 16 (ELU) -> 1 ----------------

__global__ void mlp_kernel(const float* __restrict__ h, const float* __restrict__ Wo1,
                           const float* __restrict__ bo1, const float* __restrict__ Wo2,
                           const float* __restrict__ bo2, float* __restrict__ out, int n) {
    int i = blockIdx.x * blockDim.x + threadIdx.x;
    if (i >= n) return;
    const float* hv = h + (size_t)i * HID;
    float hl[HID];
#pragma unroll
    for (int k = 0; k < HID; ++k) hl[k] = hv[k];
    float acc = bo2[0];
#pragma unroll
    for (int j = 0; j < 16; ++j) {
        float s = bo1[j];
#pragma unroll
        for (int k = 0; k < HID; ++k) s += hl[k] * Wo1[k * 16 + j];
        float e = (s > 0.0f) ? s : (expf(s) - 1.0f);   // ELU, alpha=1
        acc += e * Wo2[j];
    }
    out[i] = acc;
}

// ---------------- driver ----------------

extern "C" void kernel_launch(void* const* d_in, const int* in_sizes, int n_in,
                              void* d_out, int out_size, void* d_ws, size_t ws_size,
                              hipStream_t stream) {
    const float* x   = (const float*)d_in[0];
    const int*   ei  = (const int*)  d_in[1];
    // d_in[2] = batch (unused: single graph, identity for this output)
    const float* W1  = (const float*)d_in[3];
    const float* b1  = (const float*)d_in[4];
    const float* W2  = (const float*)d_in[5];
    const float* b2  = (const float*)d_in[6];
    const float* Wo1 = (const float*)d_in[7];
    const float* bo1 = (const float*)d_in[8];
    const float* Wo2 = (const float*)d_in[9];
    const float* bo2 = (const float*)d_in[10];

    int n = in_sizes[0] / IN_DIM;   // 100000
    int E = in_sizes[1] / 2;        // 3200000
    const int* src = ei;            // edge_index[0]
    const int* dst = ei + E;        // edge_index[1]

    float* ws   = (float*)d_ws;
    float* dinv = ws;
    float* bufA = ws + n;                       // xw / h1  (n x 32)
    float* bufB = bufA + (size_t)n * HID;       // xw2 / h2 (n x 32)
    float* agg  = bufB + (size_t)n * HID;       // scatter accumulator (n x 32)
    float* out  = (float*)d_out;

    const int B = 256;
    int nFeat   = n * HID;
    int tiles   = ((n + 15) / 16) * 2;          // 16x16 output tiles (2 N-tiles of 16)
    int gemmBlk = (tiles + 7) / 8;              // 8 waves per 256-thread block
    int scatT   = E * 4;                        // 4 threads per edge

    // degree + symmetric norm (self-loops folded into +1)
    hipMemsetAsync(dinv, 0, (size_t)n * sizeof(float), stream);
    hipMemsetAsync(agg, 0, (size_t)nFeat * sizeof(float), stream);
    deg_kernel  <<<(E + B - 1) / B, B, 0, stream>>>(dst, dinv, E);
    rsqrt_kernel<<<(n + B - 1) / B, B, 0, stream>>>(dinv, n);

    // layer 1: WMMA transform -> edge scatter -> self-loop + bias + ReLU
    gemm_x_w1      <<<gemmBlk, B, 0, stream>>>(x, W1, bufA, n);
    scatter_kernel <<<(scatT + B - 1) / B, B, 0, stream>>>(src, dst, dinv, bufA, agg, E);
    finalize_kernel<<<(nFeat + B - 1) / B, B, 0, stream>>>(agg, bufA, dinv, b1, bufA, n);

    // layer 2
    hipMemsetAsync(agg, 0, (size_t)nFeat * sizeof(float), stream);
    gemm_h_w       <<<gemmBlk, B, 0, stream>>>(bufA, W2, bufB, n);
    scatter_kernel <<<(scatT + B - 1) / B, B, 0, stream>>>(src, dst, dinv, bufB, agg, E);
    finalize_kernel<<<(nFeat + B - 1) / B, B, 0, stream>>>(agg, bufB, dinv, b2, bufB, n);

    // output MLP
    mlp_kernel<<<(n + B - 1) / B, B, 0, stream>>>(bufB, Wo1, bo1, Wo2, bo2, out, n);
}